// GATRefiner_18786186952807
// MI455X (gfx1250) — compile-verified
//
#include <hip/hip_runtime.h>
#include <hip/hip_bf16.h>
#include <math.h>

typedef float v2f __attribute__((ext_vector_type(2)));
typedef float v8f __attribute__((ext_vector_type(8)));

#define SLOPE 0.2f
#define KIN 128
#define LPITCH 132   // 128 + 4 pad: lane r hits bank (r*132)%64 = 4r -> conflict-free

__device__ __forceinline__ unsigned f2ord(float f) {
    unsigned u = __float_as_uint(f);
    return (u & 0x80000000u) ? ~u : (u | 0x80000000u);
}
__device__ __forceinline__ float ord2f(unsigned u) {
    return (u & 0x80000000u) ? __uint_as_float(u & 0x7fffffffu) : __uint_as_float(~u);
}

__global__ void k_zero(float* __restrict__ p, long n) {
    long i = (long)blockIdx.x * blockDim.x + threadIdx.x;
    long stride = (long)gridDim.x * blockDim.x;
    for (; i < n; i += stride) p[i] = 0.0f;
}

// D[n_rows x k_out] = A[n_rows x 128] @ B[128 x k_out], row-major, f32 WMMA 16x16x4.
// block = (k_out/16) waves; each wave owns one 16x16 output tile of the 16-row strip.
// A strip staged into LDS with async-copy (GLOBAL_LOAD_ASYNC_TO_LDS_B128, ASYNCcnt).
__global__ void k_gemm_wmma(const float* __restrict__ A, const float* __restrict__ B,
                            float* __restrict__ D, int n_rows, int k_out) {
    __shared__ float lds[16 * LPITCH];
    const int rowbase = blockIdx.x * 16;
    const bool full_tile = (rowbase + 16) <= n_rows;

    if (full_tile) {
        // 16 rows x 32 B128-chunks; chunk count (512) divisible by blockDim (256/64)
        for (int i = threadIdx.x; i < 16 * (KIN / 4); i += blockDim.x) {
            int r  = i >> 5;            // row in strip
            int c4 = (i & 31) << 2;     // starting float column of this 16B chunk
            unsigned lds_off = (unsigned)(size_t)(&lds[r * LPITCH + c4]);
            unsigned long long ga =
                (unsigned long long)(const void*)(A + (size_t)(rowbase + r) * KIN + c4);
            asm volatile("global_load_async_to_lds_b128 %0, %1, off"
                         :: "v"(lds_off), "v"(ga) : "memory");
        }
        asm volatile("s_wait_asynccnt 0x0" ::: "memory");
    } else {
        for (int i = threadIdx.x; i < 16 * KIN; i += blockDim.x) {
            int r = i >> 7, c = i & 127;
            int g = rowbase + r;
            lds[r * LPITCH + c] = (g < n_rows) ? A[(size_t)g * KIN + c] : 0.0f;
        }
    }
    __syncthreads();

    const int lane = threadIdx.x & 31;
    const int wave = threadIdx.x >> 5;
    const int col  = wave * 16 + (lane & 15);
    const int hi   = lane >> 4;     // 0: K pair {0,1}; 1: K pair {2,3}  (ISA A/B layout)
    const int ar   = lane & 15;

    v8f acc = {};
    for (int k = 0; k < KIN; k += 4) {
        v2f a, b;
        a.x = lds[ar * LPITCH + k + 2 * hi + 0];
        a.y = lds[ar * LPITCH + k + 2 * hi + 1];
        b.x = B[(k + 2 * hi + 0) * k_out + col];
        b.y = B[(k + 2 * hi + 1) * k_out + col];
        // (neg_a, A, neg_b, B, c_mod, C, reuse_a, reuse_b)
        acc = __builtin_amdgcn_wmma_f32_16x16x4_f32(false, a, false, b, (short)0, acc,
                                                    false, false);
    }

    // C/D layout: VGPR r -> M=r (lanes<16), M=r+8 (lanes>=16)
    if (full_tile) {
#pragma unroll
        for (int r = 0; r < 8; ++r) {
            int row = rowbase + r + 8 * hi;
            D[(size_t)row * k_out + col] = acc[r];
        }
    } else {
#pragma unroll
        for (int r = 0; r < 8; ++r) {
            int row = rowbase + r + 8 * hi;
            if (row < n_rows) D[(size_t)row * k_out + col] = acc[r];
        }
    }
}

// per-(node,head) attention logits: als = <h[n,h,:], a_src[h,:]>, ald likewise
__global__ void k_al(const float* __restrict__ h, const float* __restrict__ a_src,
                     const float* __restrict__ a_dst, float* __restrict__ als,
                     float* __restrict__ ald, int n, int heads, int cdim) {
    int idx = blockIdx.x * blockDim.x + threadIdx.x;
    if (idx >= n * heads) return;
    int node = idx / heads, hd = idx % heads;
    const float* hp = h + (long)node * heads * cdim + (long)hd * cdim;
    float s = 0.f, d = 0.f;
    for (int c = 0; c < cdim; ++c) {
        float v = hp[c];
        s += v * a_src[hd * cdim + c];
        d += v * a_dst[hd * cdim + c];
    }
    als[idx] = s;
    ald[idx] = d;
}

// pass 1: e = leakyrelu(als[src]+ald[dst]); segment max via ordered-uint atomicMax
__global__ void k_edge_max(const int* __restrict__ src, const int* __restrict__ dst,
                           int E, int Et, const float* __restrict__ als,
                           const float* __restrict__ ald, unsigned* __restrict__ emax,
                           float* __restrict__ ev, int heads) {
    long idx = (long)blockIdx.x * blockDim.x + threadIdx.x;
    if (idx >= (long)Et * heads) return;
    int e = (int)(idx / heads), hd = (int)(idx % heads);
    int s, d;
    if (e < E) { s = src[e]; d = dst[e]; } else { s = d = e - E; }
    float v = als[s * heads + hd] + ald[d * heads + hd];
    v = v > 0.f ? v : SLOPE * v;
    ev[idx] = v;
    atomicMax(&emax[d * heads + hd], f2ord(v));
}

// pass 2: ev = exp(e - emax[dst]); segment sum of denominators
__global__ void k_edge_exp(const int* __restrict__ src, const int* __restrict__ dst,
                           int E, int Et, const unsigned* __restrict__ emax,
                           float* __restrict__ denom, float* __restrict__ ev, int heads) {
    long idx = (long)blockIdx.x * blockDim.x + threadIdx.x;
    if (idx >= (long)Et * heads) return;
    int e = (int)(idx / heads), hd = (int)(idx % heads);
    int d = (e < E) ? dst[e] : (e - E);
    float ex = expf(ev[idx] - ord2f(emax[d * heads + hd]));
    ev[idx] = ex;
    atomicAdd(&denom[d * heads + hd], ex);
}

// pass 3: one block per edge (heads*cdim threads): agg[dst] += h[src] * alpha
__global__ void k_edge_agg(const int* __restrict__ src, const int* __restrict__ dst,
                           int E, const float* __restrict__ ev,
                           const float* __restrict__ denom, const float* __restrict__ hpre,
                           float* __restrict__ agg, int heads, int cdim) {
    int e = blockIdx.x;
    int t = threadIdx.x;          // width = heads*cdim
    int hd = t / cdim;
    int s, d;
    if (e < E) { s = src[e]; d = dst[e]; } else { s = d = e - E; }
    float alpha = ev[(long)e * heads + hd] / (denom[d * heads + hd] + 1e-16f);
    int width = heads * cdim;
    atomicAdd(&agg[(long)d * width + t], hpre[(long)s * width + t] * alpha);
}

// epilogue: h = (do_elu ? elu : id)(h + b)
__global__ void k_bias_act(float* __restrict__ h, const float* __restrict__ b,
                           long total, int width, int do_elu) {
    long i = (long)blockIdx.x * blockDim.x + threadIdx.x;
    long stride = (long)gridDim.x * blockDim.x;
    for (; i < total; i += stride) {
        float v = h[i] + b[(int)(i % width)];
        if (do_elu) v = v > 0.f ? v : expm1f(v);
        h[i] = v;
    }
}

// final head: one wave per node. node_output = h3 @ Wc + bc; link_embedding = h3
__global__ void k_head(const float* __restrict__ h3, const float* __restrict__ Wc,
                       const float* __restrict__ bc, float* __restrict__ out_node,
                       float* __restrict__ out_link, int n) {
    int wave = threadIdx.x >> 5;
    int lane = threadIdx.x & 31;
    int node = blockIdx.x * (blockDim.x >> 5) + wave;
    if (node >= n) return;
    float v = h3[(long)node * 32 + lane];
    out_link[(long)node * 32 + lane] = v;
    float p0 = v * Wc[lane * 2 + 0];
    float p1 = v * Wc[lane * 2 + 1];
#pragma unroll
    for (int off = 16; off; off >>= 1) {
        p0 += __shfl_xor(p0, off, 32);
        p1 += __shfl_xor(p1, off, 32);
    }
    if (lane == 0) {
        out_node[(long)node * 2 + 0] = p0 + bc[0];
        out_node[(long)node * 2 + 1] = p1 + bc[1];
    }
}

static inline int ceil_div(long a, long b) { return (int)((a + b - 1) / b); }

extern "C" void kernel_launch(void* const* d_in, const int* in_sizes, int n_in,
                              void* d_out, int out_size, void* d_ws, size_t ws_size,
                              hipStream_t stream) {
    const float* x      = (const float*)d_in[0];
    const int*   ei     = (const int*)d_in[1];
    const float* W1     = (const float*)d_in[2];
    const float* a1_src = (const float*)d_in[3];
    const float* a1_dst = (const float*)d_in[4];
    const float* b1     = (const float*)d_in[5];
    const float* W2     = (const float*)d_in[6];
    const float* a2_src = (const float*)d_in[7];
    const float* a2_dst = (const float*)d_in[8];
    const float* b2     = (const float*)d_in[9];
    const float* W3     = (const float*)d_in[10];
    const float* a3_src = (const float*)d_in[11];
    const float* a3_dst = (const float*)d_in[12];
    const float* b3     = (const float*)d_in[13];
    const float* Wc     = (const float*)d_in[14];
    const float* bc     = (const float*)d_in[15];

    const int N  = in_sizes[0] / 128;
    const int E  = in_sizes[1] / 2;
    const int Et = E + N;
    const int* src = ei;
    const int* dst = ei + E;

    float* ws   = (float*)d_ws;
    float* buf0 = ws;                              // N*128  (pre-aggregation h)
    float* buf1 = buf0 + (size_t)N * 128;          // N*128  (aggregated layer output)
    float* als  = buf1 + (size_t)N * 128;          // N*4
    float* ald  = als + (size_t)N * 4;             // N*4
    unsigned* emax = (unsigned*)(ald + (size_t)N * 4); // N*4
    float* denom = (float*)emax + (size_t)N * 4;   // N*4
    float* ev    = denom + (size_t)N * 4;          // Et*4 (per-edge logits/weights)

    auto layer = [&](const float* in, const float* W, const float* asrc,
                     const float* adst, const float* b, int heads, int kout,
                     int do_elu, float* hpre, float* agg) {
        const int width = heads * 32;              // == kout for all three layers
        k_gemm_wmma<<<ceil_div(N, 16), (kout / 16) * 32, 0, stream>>>(in, W, hpre, N, kout);
        k_al<<<ceil_div((long)N * heads, 256), 256, 0, stream>>>(hpre, asrc, adst,
                                                                 als, ald, N, heads, 32);
        k_zero<<<1024, 256, 0, stream>>>((float*)emax, (long)N * heads);
        k_zero<<<1024, 256, 0, stream>>>(denom, (long)N * heads);
        k_zero<<<4096, 256, 0, stream>>>(agg, (long)N * width);
        k_edge_max<<<ceil_div((long)Et * heads, 256), 256, 0, stream>>>(
            src, dst, E, Et, als, ald, emax, ev, heads);
        k_edge_exp<<<ceil_div((long)Et * heads, 256), 256, 0, stream>>>(
            src, dst, E, Et, emax, denom, ev, heads);
        k_edge_agg<<<Et, width, 0, stream>>>(src, dst, E, ev, denom, hpre, agg, heads, 32);
        k_bias_act<<<ceil_div((long)N * width, 256), 256, 0, stream>>>(
            agg, b, (long)N * width, width, do_elu);
    };

    // conv1: IN=128 -> H*C=128, elu
    layer(x,    W1, a1_src, a1_dst, b1, 4, 128, 1, buf0, buf1);
    // conv2: 128 -> 128, elu
    layer(buf1, W2, a2_src, a2_dst, b2, 4, 128, 1, buf0, buf1);
    // conv3: 128 -> 32, heads=1, no elu
    layer(buf1, W3, a3_src, a3_dst, b3, 1, 32, 0, buf0, buf1);

    // head: node_output [N,2] then link_embedding [N,32]
    float* out_node = (float*)d_out;
    float* out_link = out_node + (size_t)N * 2;
    k_head<<<ceil_div(N, 8), 256, 0, stream>>>(buf1, Wc, bc, out_node, out_link, N);
}